// CLCRNCell_53085795778848
// MI455X (gfx1250) — compile-verified
//
#include <hip/hip_runtime.h>
#include <math.h>

// Problem constants (from reference)
#define B_      4
#define N_      10000
#define F_IN_   64
#define UNITS_  64
#define E_      160000
#define C_      128            // F_IN + UNITS
#define M_      (B_ * N_)      // 40000 GEMM rows
#define MV_     3
#define KTOT_   (MV_ * C_)     // 384
#define LDSW_   388            // padded LDS row stride (floats); 388 % 64 = 4
                               // -> conflict-free b64 A-fragment reads

typedef __attribute__((ext_vector_type(2))) float v2f;
typedef __attribute__((ext_vector_type(8))) float v8f;

// ---------------- elementwise helpers ----------------

__global__ void k_zero(float* __restrict__ p, int n) {
  int i = blockIdx.x * blockDim.x + threadIdx.x;
  if (i < n) p[i] = 0.0f;
}

// v0 = concat(inputs, hx) along channels -> (M_, 128)
__global__ void k_concat_ru(const float* __restrict__ in,
                            const float* __restrict__ hx,
                            float* __restrict__ v0) {
  int idx = blockIdx.x * blockDim.x + threadIdx.x;      // over M_*C_
  if (idx >= M_ * C_) return;
  int m  = idx >> 7;
  int ch = idx & (C_ - 1);
  float v = (ch < F_IN_) ? in[(size_t)m * F_IN_ + ch]
                         : hx[(size_t)m * UNITS_ + (ch - F_IN_)];
  v0[idx] = v;
}

// v0 = concat(inputs, r * hx); r = ru[:, 0:64]
__global__ void k_concat_c(const float* __restrict__ in,
                           const float* __restrict__ hx,
                           const float* __restrict__ ru,
                           float* __restrict__ v0) {
  int idx = blockIdx.x * blockDim.x + threadIdx.x;
  if (idx >= M_ * C_) return;
  int m  = idx >> 7;
  int ch = idx & (C_ - 1);
  float v;
  if (ch < F_IN_) {
    v = in[(size_t)m * F_IN_ + ch];
  } else {
    int u = ch - F_IN_;
    float r = ru[(size_t)m * (2 * UNITS_) + u];         // r = first half of ru
    v = r * hx[(size_t)m * UNITS_ + u];
  }
  v0[idx] = v;
}

// ---------------- sparse propagation: y[:,dst,:] += ker[e] * x[:,src,:] -----
// One lane handles 4 channels of one edge for all 4 batches (float4 gather,
// 4 L2 atomics). Working set per (batch,view) is 5.1MB -> L2 resident.
__global__ void k_scatter(const float* __restrict__ x,
                          float* __restrict__ y,
                          const int* __restrict__ src,
                          const int* __restrict__ dst,
                          const float* __restrict__ ker) {
  int idx = blockIdx.x * blockDim.x + threadIdx.x;      // over E_*32
  if (idx >= E_ * 32) return;
  int e = idx >> 5;
  int q = (idx & 31) << 2;                              // channel base (0..124)
  // speculative prefetch of edge metadata ahead (global_prefetch_b8)
  __builtin_prefetch(&src[e + 4096], 0, 0);
  __builtin_prefetch(&dst[e + 4096], 0, 0);
  int   s = src[e];
  int   d = dst[e];
  float k = ker[e];
#pragma unroll
  for (int b = 0; b < B_; ++b) {
    const float4 v = *(const float4*)(x + ((size_t)(b * N_ + s)) * C_ + q);
    float* yp = y + ((size_t)(b * N_ + d)) * C_ + q;
    atomicAdd(yp + 0, k * v.x);
    atomicAdd(yp + 1, k * v.y);
    atomicAdd(yp + 2, k * v.z);
    atomicAdd(yp + 3, k * v.w);
  }
}

// ---------------- fp32 WMMA GEMM core ----------------
// A tile (16 rows x 384 K) staged once per block into LDS (24KB, padded rows),
// then every wave feeds its WMMA A fragments from LDS (ds_load_b64,
// conflict-free via LDSW_ padding) instead of 8x-redundant global loads.
// A 16x4 f32 frag: lanes 0-15 rows m0+lane K={k0,k0+1}; lanes 16-31 K={k0+2,k0+3}.
// B 4x16 f32 frag mirrors per-column. C/D: VGPR r <-> M=r (+8 for hi lanes).

__device__ inline void fill_As(float* __restrict__ As,
                               const float* __restrict__ v0,
                               const float* __restrict__ v1,
                               const float* __restrict__ v2,
                               int m0, int tid, int nthreads) {
  // 16 rows x 384 floats = 1536 float4 chunks; views are contiguous 128-chunks
  for (int i = tid; i < 16 * (KTOT_ / 4); i += nthreads) {
    int row  = i / (KTOT_ / 4);
    int kk4  = (i % (KTOT_ / 4)) * 4;                   // 0..380
    int view = kk4 >> 7;
    int ch   = kk4 & (C_ - 1);
    const float* srcp = (view == 0 ? v0 : (view == 1 ? v1 : v2))
                        + ((size_t)(m0 + row)) * C_ + ch;
    *(float4*)(As + row * LDSW_ + kk4) = *(const float4*)srcp;  // ds_store_b128
  }
  __syncthreads();
}

__device__ inline v8f wmma_tile_lds(const float* __restrict__ As,
                                    const float* __restrict__ W, int nout,
                                    int n0, int lane) {
  const int arow = lane & 15;
  const int koff = (lane >> 4) << 1;                    // 0 or 2
  const int col  = n0 + (lane & 15);
  v8f acc = {};
#pragma unroll 4
  for (int kb = 0; kb < KTOT_ / 4; ++kb) {
    const int k0 = kb * 4 + koff;
    v2f a = *(const v2f*)(As + arow * LDSW_ + k0);      // ds_load_b64
    v2f b;
    b.x = W[(size_t)k0 * nout + col];
    b.y = W[(size_t)(k0 + 1) * nout + col];
    acc = __builtin_amdgcn_wmma_f32_16x16x4_f32(
        /*neg_a=*/false, a, /*neg_b=*/false, b,
        /*c_mod=*/(short)0, acc, /*reuse_a=*/false, /*reuse_b=*/false);
  }
  return acc;
}

// ru = sigmoid(H @ W_ru + b_ru), NOUT = 128. Block = 8 waves, wave w -> n0=16w.
__global__ void k_gemm_ru(const float* __restrict__ v0,
                          const float* __restrict__ v1,
                          const float* __restrict__ v2,
                          const float* __restrict__ W,
                          const float* __restrict__ bias,
                          float* __restrict__ ru) {
  __shared__ float As[16 * LDSW_];
  const int lane = threadIdx.x & 31;
  const int wave = threadIdx.x >> 5;
  const int m0 = blockIdx.x * 16;
  const int n0 = wave * 16;
  fill_As(As, v0, v1, v2, m0, threadIdx.x, 256);
  v8f acc = wmma_tile_lds(As, W, 2 * UNITS_, n0, lane);
  const int col   = n0 + (lane & 15);
  const int mbase = m0 + ((lane >> 4) << 3);
  const float bv  = bias[col];
#pragma unroll
  for (int r = 0; r < 8; ++r) {
    int m = mbase + r;
    float x = acc[r] + bv;
    ru[(size_t)m * (2 * UNITS_) + col] = 1.0f / (1.0f + __expf(-x));
  }
}

// c = tanh(H @ W_c + b_c), NOUT = 64; fused GRU output epilogue.
// Block = 4 waves (n0 = 16w covers 64 cols).
__global__ void k_gemm_c(const float* __restrict__ v0,
                         const float* __restrict__ v1,
                         const float* __restrict__ v2,
                         const float* __restrict__ W,
                         const float* __restrict__ bias,
                         const float* __restrict__ ru,
                         const float* __restrict__ hx,
                         float* __restrict__ out) {
  __shared__ float As[16 * LDSW_];
  const int lane = threadIdx.x & 31;
  const int wave = threadIdx.x >> 5;
  const int m0 = blockIdx.x * 16;
  const int n0 = wave * 16;
  fill_As(As, v0, v1, v2, m0, threadIdx.x, 128);
  v8f acc = wmma_tile_lds(As, W, UNITS_, n0, lane);
  const int col   = n0 + (lane & 15);
  const int mbase = m0 + ((lane >> 4) << 3);
  const float bv  = bias[col];
#pragma unroll
  for (int r = 0; r < 8; ++r) {
    int m = mbase + r;
    float c = tanhf(acc[r] + bv);
    float u = ru[(size_t)m * (2 * UNITS_) + UNITS_ + col];  // u = second half
    float h = hx[(size_t)m * UNITS_ + col];
    out[(size_t)m * UNITS_ + col] = u * h + (1.0f - u) * c;
  }
}

// ---------------- launch ----------------
extern "C" void kernel_launch(void* const* d_in, const int* in_sizes, int n_in,
                              void* d_out, int out_size, void* d_ws, size_t ws_size,
                              hipStream_t stream) {
  const float* inputs = (const float*)d_in[0];
  const float* hx     = (const float*)d_in[1];
  const int*   sidx   = (const int*)  d_in[2];   // (2, E): row0=src, row1=dst
  const float* ker    = (const float*)d_in[3];
  const float* W_ru   = (const float*)d_in[4];
  const float* b_ru   = (const float*)d_in[5];
  const float* W_c    = (const float*)d_in[6];
  const float* b_c    = (const float*)d_in[7];
  float* out = (float*)d_out;

  float* ws = (float*)d_ws;
  const size_t VSZ = (size_t)M_ * C_;            // 5.12M floats per view
  float* v0 = ws;
  float* v1 = ws + VSZ;
  float* v2 = ws + 2 * VSZ;
  float* ru = ws + 3 * VSZ;                      // (M_, 128)

  const int* src = sidx;
  const int* dst = sidx + E_;

  const int T = 256;
  const int gridMC   = (M_ * C_ + T - 1) / T;    // 20000 blocks
  const int gridScat = (E_ * 32 + T - 1) / T;    // 20000 blocks
  const int gridGemm = M_ / 16;                  // 2500 tiles

  // ---- gate conv: views over concat(inputs, hx) ----
  k_concat_ru<<<gridMC, T, 0, stream>>>(inputs, hx, v0);
  k_zero<<<gridMC, T, 0, stream>>>(v1, M_ * C_);
  k_scatter<<<gridScat, T, 0, stream>>>(v0, v1, src, dst, ker);
  k_zero<<<gridMC, T, 0, stream>>>(v2, M_ * C_);
  k_scatter<<<gridScat, T, 0, stream>>>(v1, v2, src, dst, ker);
  k_gemm_ru<<<gridGemm, 256, 0, stream>>>(v0, v1, v2, W_ru, b_ru, ru);

  // ---- candidate conv: views over concat(inputs, r*hx) ----
  k_concat_c<<<gridMC, T, 0, stream>>>(inputs, hx, ru, v0);
  k_zero<<<gridMC, T, 0, stream>>>(v1, M_ * C_);
  k_scatter<<<gridScat, T, 0, stream>>>(v0, v1, src, dst, ker);
  k_zero<<<gridMC, T, 0, stream>>>(v2, M_ * C_);
  k_scatter<<<gridScat, T, 0, stream>>>(v1, v2, src, dst, ker);
  k_gemm_c<<<gridGemm, 128, 0, stream>>>(v0, v1, v2, W_c, b_c, ru, hx, out);
}